// SamEncoderEmbeddingsSegmentsEncoder_25374666785147
// MI455X (gfx1250) — compile-verified
//
#include <hip/hip_runtime.h>

// ---------------------------------------------------------------------------
// SamEncoderEmbeddingsSegmentsEncoder for MI455X (gfx1250, wave32)
//   S=128 segments, E=16 images, C=256 channels, H=1024 mask, h=64, r=16
// Phase 1 (bandwidth bound, ~512MB @ 23.3TB/s): 16x16 block-sum of masks,
//          threshold -> sel (S,4096) f32 {0,1}, den[s] = popcount as f32.
// Phase 2: bucket segments by image id (A matrix shared across WMMA columns).
// Phase 3: per-image GEMM  num = sam[e] (256x4096) x selcols (4096 x n_e)
//          using V_WMMA_F32_16X16X4_F32 (f32 in / f32 accum, matches ref
//          precision), 4 waves split K, LDS tree-reduce, divide by den,
//          scatter into d_out.
// Phase 4: pass-through scalar outputs.
// ---------------------------------------------------------------------------

typedef __attribute__((ext_vector_type(2))) float v2f;
typedef __attribute__((ext_vector_type(8))) float v8f;

constexpr int SEG   = 128;     // segments
constexpr int NIMG  = 16;      // encoder images
constexpr int NCH   = 256;     // channels
constexpr int HFULL = 1024;    // mask resolution
constexpr int HC    = 64;      // coarse grid (h)
constexpr int KHW   = HC * HC; // 4096 = flattened h*w
constexpr float MIN_PIX = 50.0f;

// ---------------------------------------------------------------------------
// Phase 2 + zero-init: deterministic single-lane bucketing of segments by
// image id; also zeroes the den accumulators used by the mask kernel.
__global__ void k_bucket(const int* __restrict__ ids,
                         float* __restrict__ den,
                         int* __restrict__ cnt,
                         int* __restrict__ seglist) {
  if (threadIdx.x == 0 && blockIdx.x == 0) {
    for (int e = 0; e < NIMG; ++e) cnt[e] = 0;
    for (int s = 0; s < SEG; ++s) den[s] = 0.0f;
    for (int s = 0; s < SEG; ++s) {
      int e = ids[s];
      seglist[e * SEG + cnt[e]] = s;
      cnt[e]++;
    }
  }
}

// ---------------------------------------------------------------------------
// Phase 1: block = (segment s, coarse row i). 256 threads stream 16 rows of
// 1024 floats with float4 loads (fully coalesced). Thread t covers columns
// [4t, 4t+4) -> cell j = t>>2 is finished by a 4-lane shfl_xor reduction
// (lanes 4j..4j+3 lie in one wave32). Exact: sums are small integers in f32.
__global__ void k_masksum(const float* __restrict__ masks,
                          float* __restrict__ sel,
                          float* __restrict__ den) {
  const int i = blockIdx.x;   // coarse row 0..63
  const int s = blockIdx.y;   // segment
  const int t = threadIdx.x;  // 0..255

  const float4* base =
      (const float4*)(masks + (size_t)s * HFULL * HFULL + (size_t)(i * 16) * HFULL) + t;

  if (i < HC - 1) {
    // speculative prefetch of the next 16-row band (global_prefetch_b8)
    __builtin_prefetch(base + 16 * (HFULL / 4), 0, 0);
  }

  float sum = 0.0f;
#pragma unroll
  for (int rr = 0; rr < 16; ++rr) {
    float4 v = base[rr * (HFULL / 4)];
    sum += v.x + v.y + v.z + v.w;
  }
  // reduce groups of 4 lanes -> one 16x16 cell
  sum += __shfl_xor(sum, 1, 32);
  sum += __shfl_xor(sum, 2, 32);

  __shared__ float cacc;
  if (t == 0) cacc = 0.0f;
  __syncthreads();

  if ((t & 3) == 0) {
    const int j = t >> 2;  // cell column 0..63
    const float sv = (sum >= MIN_PIX) ? 1.0f : 0.0f;
    sel[(size_t)s * KHW + i * HC + j] = sv;
    atomicAdd(&cacc, sv);          // ds_add_f32
  }
  __syncthreads();
  if (t == 0) atomicAdd(&den[s], cacc);  // one global f32 atomic per block
}

// ---------------------------------------------------------------------------
// Phase 3: per-image GEMM tile via V_WMMA_F32_16X16X4_F32.
//   block = (nt segment-tile, mt channel-tile, e image), 128 threads = 4 waves
//   D[m,n] += A[m,kk] * B[kk,n] with
//     A[m,kk] = sam[e, mt*16+m, k0+kk]   (16x4 f32, lane = {M=lane%16, K pair})
//     B[kk,n] = sel[seg(n),     k0+kk]   (4x16 f32, lane = {N=lane%16, K pair})
//   Each wave covers a K slice of 1024; partial tiles reduced through LDS.
__global__ void k_gemm(const float* __restrict__ sam,
                       const float* __restrict__ sel,
                       const float* __restrict__ den,
                       const int* __restrict__ cnt,
                       const int* __restrict__ seglist,
                       float* __restrict__ outEmb) {
  const int nt = blockIdx.x;   // segment column tile within image bucket
  const int mt = blockIdx.y;   // channel tile (16 channels)
  const int e  = blockIdx.z;   // image

  const int nSeg = cnt[e];
  if (nt * 16 >= nSeg) return;   // block-uniform: EXEC stays all-ones below

  const int lane = threadIdx.x & 31;
  const int wave = threadIdx.x >> 5;   // 0..3
  const int lo   = lane & 15;
  const int hi   = lane >> 4;          // K sub-pair select (ISA 16x4 layout)

  // A operand: row = channel (mt*16 + lo), lane-half selects K offset {0,2}
  const float* Aptr = sam + ((size_t)e * NCH + mt * 16 + lo) * KHW + 2 * hi;

  // B operand: column lo -> segment of this image's bucket (zero-pad tail)
  const int   colIdx = nt * 16 + lo;
  const int   valid  = colIdx < nSeg;
  const int   seg    = valid ? seglist[e * SEG + colIdx] : 0;
  const float bscale = valid ? 1.0f : 0.0f;
  const float* Bptr  = sel + (size_t)seg * KHW + 2 * hi;

  v8f acc = {0.f, 0.f, 0.f, 0.f, 0.f, 0.f, 0.f, 0.f};
  const int kBeg = wave * (KHW / 4);   // 4 waves x 1024 K each
#pragma unroll 4
  for (int k = kBeg; k < kBeg + KHW / 4; k += 4) {
    v2f a = *(const v2f*)(Aptr + k);
    v2f b = *(const v2f*)(Bptr + k);
    b *= bscale;
    // D = A(16x4 f32) x B(4x16 f32) + C, full f32 accumulate
    acc = __builtin_amdgcn_wmma_f32_16x16x4_f32(
        /*neg_a=*/false, a, /*neg_b=*/false, b,
        /*c_mod=*/(short)0, acc, /*reuse_a=*/false, /*reuse_b=*/false);
  }

  // cross-wave K reduction through LDS
  __shared__ float red[4][256];
#pragma unroll
  for (int v = 0; v < 8; ++v) red[wave][lane * 8 + v] = acc[v];
  __syncthreads();

  if (wave == 0) {
#pragma unroll
    for (int v = 0; v < 8; ++v) {
      const float r = red[0][lane * 8 + v] + red[1][lane * 8 + v] +
                      red[2][lane * 8 + v] + red[3][lane * 8 + v];
      // C/D layout: lane 0-15 -> M = v, lane 16-31 -> M = v + 8; N = lane%16
      const int m = v + 8 * hi;
      const int c = mt * 16 + m;
      if (valid) {
        outEmb[(size_t)seg * NCH + c] = r / den[seg];
      }
    }
  }
}

// ---------------------------------------------------------------------------
// Phase 4: pass-through outputs (int -> float32 value conversion; bool -> 0).
// d_out layout (flat, return order): image_ids[128] | relative_segment_ids[128]
//   | is_latent_tokens[128] | segments_embeddings[128*256] | coords[128*4]
__global__ void k_scalars(const int* __restrict__ ids,
                          const int* __restrict__ rel,
                          const float* __restrict__ coords,
                          float* __restrict__ out) {
  const int t = threadIdx.x + blockIdx.x * blockDim.x;  // 0..511
  if (t < SEG) {
    out[t]           = (float)ids[t];
    out[SEG + t]     = (float)rel[t];
    out[2 * SEG + t] = 0.0f;            // is_latent_tokens = False
  }
  if (t < 4 * SEG) {
    out[3 * SEG + SEG * NCH + t] = coords[t];
  }
}

// ---------------------------------------------------------------------------
extern "C" void kernel_launch(void* const* d_in, const int* in_sizes, int n_in,
                              void* d_out, int out_size, void* d_ws, size_t ws_size,
                              hipStream_t stream) {
  (void)in_sizes; (void)n_in; (void)out_size; (void)ws_size;

  const float* masks  = (const float*)d_in[0];  // (S,1,H,H)    f32
  const float* sam    = (const float*)d_in[1];  // (E,1,C,h,h)  f32
  const int*   ids    = (const int*)d_in[2];    // (S,)         i32
  const int*   rel    = (const int*)d_in[3];    // (S,)         i32
  const float* coords = (const float*)d_in[4];  // (S,4)        f32
  float* out = (float*)d_out;

  // workspace: sel (S*4096 f32 = 2MB) | den (S f32) | cnt (E i32) | seglist
  float* sel     = (float*)d_ws;
  float* den     = sel + (size_t)SEG * KHW;
  int*   cnt     = (int*)(den + SEG);
  int*   seglist = cnt + NIMG;

  hipLaunchKernelGGL(k_bucket, dim3(1), dim3(32), 0, stream,
                     ids, den, cnt, seglist);
  hipLaunchKernelGGL(k_masksum, dim3(HC, SEG), dim3(256), 0, stream,
                     masks, sel, den);
  hipLaunchKernelGGL(k_gemm, dim3(SEG / 16, NCH / 16, NIMG), dim3(128), 0, stream,
                     sam, sel, den, cnt, seglist, out + 3 * SEG);
  hipLaunchKernelGGL(k_scalars, dim3(2), dim3(256), 0, stream,
                     ids, rel, coords, out);
}